// GaussianAntecedent_26792005992674
// MI455X (gfx1250) — compile-verified
//
#include <hip/hip_runtime.h>

typedef float v2f __attribute__((ext_vector_type(2)));
typedef float v8f __attribute__((ext_vector_type(8)));

#define EPS 1e-8f
// Row stride for the K-pair-interleaved W tile, in float2 units.
// 80 float2 = 160 dwords == 32 (mod 64 banks) -> the two lane-halves (which read
// adjacent kp rows) hit disjoint 32-bank groups: conflict-free B fragment loads.
#define WP_STRIDE 80

__global__ __launch_bounds__(256) void gauss_antecedent_wmma(
    const float* __restrict__ X,
    const float* __restrict__ centers,
    const float* __restrict__ sigma,
    float* __restrict__ out,
    int nrows)
{
    // Wp[kp][r] = { W[2kp][r], W[2kp+1][r] },  kp in [0,32), r in [0,64)
    // K layout: k<32 -> linear coeff -w*c ; k>=32 -> quadratic coeff 0.5*w
    __shared__ v2f   Wp[32 * WP_STRIDE];
    __shared__ float biasLds[64];

    const int tid = threadIdx.x;

    // ---- Stage W into LDS, pre-interleaved by K-pair (8B global loads + 8B DS stores) ----
    for (int idx = tid; idx < 64 * 16; idx += 256) {
        const int r = idx >> 4;       // rule
        const int j = idx & 15;       // feature pair: d = 2j, 2j+1
        const v2f sg = *reinterpret_cast<const v2f*>(sigma   + r * 32 + 2 * j);
        const v2f cn = *reinterpret_cast<const v2f*>(centers + r * 32 + 2 * j);
        v2f w;
        {
            const float s0 = sg.x + EPS, s1 = sg.y + EPS;
            w.x = 1.0f / (s0 * s0);
            w.y = 1.0f / (s1 * s1);
        }
        v2f lin, quad;
        lin.x  = -w.x * cn.x;  lin.y  = -w.y * cn.y;   // K = 2j, 2j+1
        quad.x = 0.5f * w.x;   quad.y = 0.5f * w.y;    // K = 32+2j, 33+2j -> kp = 16+j
        Wp[j * WP_STRIDE + r]        = lin;
        Wp[(16 + j) * WP_STRIDE + r] = quad;
    }
    if (tid < 64) {
        float bs = 0.0f;
        for (int d = 0; d < 32; ++d) {
            const float sd = sigma[tid * 32 + d] + EPS;
            const float w  = 1.0f / (sd * sd);
            const float c  = centers[tid * 32 + d];
            bs += 0.5f * w * c * c;
        }
        biasLds[tid] = bs;
    }
    __syncthreads();

    // ---- Per-wave tile: 16 rows x 64 rules ----
    const int lane    = tid & 31;
    const int wave    = tid >> 5;
    const int rowbase = (blockIdx.x * 8 + wave) * 16;
    if (rowbase >= nrows) return;     // wave-uniform: surviving waves keep EXEC all-ones for WMMA

    const int l    = lane & 15;       // row within tile (A) / column within r-tile (B,C,D)
    const int half = lane >> 4;

    // A fragments: 16 steps of K=4. Steps 0..7 = x (K<32), steps 8..15 = x^2 (K>=32).
    v2f a[16];
    const float* xrow = X + (size_t)(rowbase + l) * 32 + half * 2;
#pragma unroll
    for (int kk = 0; kk < 8; ++kk)
        a[kk] = *reinterpret_cast<const v2f*>(xrow + 4 * kk);   // 8B global loads
#pragma unroll
    for (int kk = 0; kk < 8; ++kk)
        a[kk + 8] = a[kk] * a[kk];

    // Single base address for all B fragments; every step uses an immediate DS offset:
    //   fragment(kk, rt) = wp[kk*2*WP_STRIDE + rt*16]   (kp = 2kk + half folded into base)
    const v2f* wp = Wp + half * WP_STRIDE + l;

    // Accumulators initialized with bias[r] so the WMMA chain yields the full exponent S.
    v8f acc[4];
#pragma unroll
    for (int rt = 0; rt < 4; ++rt) {
        const float b = biasLds[rt * 16 + l];
        acc[rt] = (v8f){b, b, b, b, b, b, b, b};
    }

    // GEMM: 16 K-steps x 4 R-tiles; rt inner so the 4 accumulator chains interleave.
#pragma unroll
    for (int kk = 0; kk < 16; ++kk) {
#pragma unroll
        for (int rt = 0; rt < 4; ++rt) {
            const v2f bf = wp[kk * (2 * WP_STRIDE) + rt * 16];  // one ds_load_b64, imm offset
            acc[rt] = __builtin_amdgcn_wmma_f32_16x16x4_f32(
                false, a[kk], false, bf, (short)0, acc[rt], false, false);
        }
    }

    // mem = exp(-S)   (underflow to 0 matches fp32 reference behavior)
#pragma unroll
    for (int rt = 0; rt < 4; ++rt)
#pragma unroll
        for (int v = 0; v < 8; ++v)
            acc[rt][v] = __expf(-acc[rt][v]);

    // Row sums over all 64 rules: in-lane over rt, then xor-reduce within each 16-lane half
    // (C-layout: lanes 0-15 hold rows m=v, lanes 16-31 hold rows m=v+8).
    float rinv[8];
#pragma unroll
    for (int v = 0; v < 8; ++v) {
        float s = acc[0][v] + acc[1][v] + acc[2][v] + acc[3][v];
        s += __shfl_xor(s, 1, 32);
        s += __shfl_xor(s, 2, 32);
        s += __shfl_xor(s, 4, 32);
        s += __shfl_xor(s, 8, 32);
        rinv[v] = 1.0f / (s + EPS);
    }

    // Store normalized memberships: out[n][r], n = rowbase + half*8 + v, r = rt*16 + l.
    // Each store instruction covers two full 64B segments (lanes 0-15 / 16-31).
    float* orow = out + (size_t)(rowbase + half * 8) * 64 + l;
#pragma unroll
    for (int v = 0; v < 8; ++v)
#pragma unroll
        for (int rt = 0; rt < 4; ++rt)
            orow[(size_t)v * 64 + rt * 16] = acc[rt][v] * rinv[v];
}

extern "C" void kernel_launch(void* const* d_in, const int* in_sizes, int n_in,
                              void* d_out, int out_size, void* d_ws, size_t ws_size,
                              hipStream_t stream) {
    const float* X       = (const float*)d_in[0];
    const float* centers = (const float*)d_in[1];
    const float* sigma   = (const float*)d_in[2];
    float* out           = (float*)d_out;

    const int n = in_sizes[0] / 32;              // N = 100000 (divisible by 16)
    const int rowsPerBlock = 8 * 16;             // 8 waves x 16 rows
    const int blocks = (n + rowsPerBlock - 1) / rowsPerBlock;

    gauss_antecedent_wmma<<<blocks, 256, 0, stream>>>(X, centers, sigma, out, n);
}